// Node_OP_18150531793353
// MI455X (gfx1250) — compile-verified
//
#include <hip/hip_runtime.h>
#include <math.h>

// MI455X (gfx1250) fused GIN layer.
// Pipeline (all on `stream`):
//  K1 merge:   out = s0*h0+s1*h1 + vn_aggr[batch];  aggr=0;  vn_temp=vn_aggr
//  K2 segsum:  vn_temp += segment_sum(out, batch)   (run-length + atomics, batch sorted)
//  K3 edges:   aggr[dst] += relu(out[src] + bond_emb sum)   (float4 per 16-lane group)
//  K4 mlp(vn): vn_out = relu(BN(relu(BN(vn_temp@vW1+vb1))@vW2+vb2))   -- WMMA f32 16x16x4
//  K5 mlp(h):  h = relu(BN(relu(BN(((1+eps)out+aggr)@W1+b1))@W2+b2))  -- WMMA f32 16x16x4
// The edge scatter (~1 GB of HBM traffic) bounds runtime; the MLPs are f32 WMMA
// (exact precision) and fused through LDS so the 16x128 intermediate never
// touches HBM.

typedef __attribute__((ext_vector_type(2))) float v2f;
typedef __attribute__((ext_vector_type(8))) float v8f;

#define D_ 64
#define BN_EPS_ 1e-5f

__device__ __forceinline__ v8f wmma_f32_k4(v2f a, v2f b, v8f c) {
  // V_WMMA_F32_16X16X4_F32: D = A(16x4,f32) x B(4x16,f32) + C(16x16,f32)
  return __builtin_amdgcn_wmma_f32_16x16x4_f32(false, a, false, b, (short)0, c,
                                               false, false);
}

// ---------------------------------------------------------------- K1: merge
__global__ __launch_bounds__(256) void k_merge(
    const float* __restrict__ h0, const float* __restrict__ h1,
    const float* __restrict__ vn0, const float* __restrict__ vn1,
    const int* __restrict__ batch,
    const float* __restrict__ mw, const float* __restrict__ mwv,
    float* __restrict__ out, float* __restrict__ aggr,
    float* __restrict__ vn_temp, int n, int g) {
  int idx = blockIdx.x * 256 + threadIdx.x;
  int total = n * D_;
  if (idx >= total) return;
  float s0 = 1.0f / (1.0f + __expf(-mw[0]));
  float s1 = 1.0f / (1.0f + __expf(-mw[1]));
  float sv0 = 1.0f / (1.0f + __expf(-mwv[0]));
  float sv1 = 1.0f / (1.0f + __expf(-mwv[1]));
  int row = idx >> 6;
  int c = idx & 63;
  int b = batch[row];
  float va = sv0 * vn0[b * D_ + c] + sv1 * vn1[b * D_ + c];
  out[idx] = s0 * h0[idx] + s1 * h1[idx] + va;
  aggr[idx] = 0.0f;
  if (idx < g * D_) vn_temp[idx] = sv0 * vn0[idx] + sv1 * vn1[idx];
}

// ------------------------------------------------------------- K2: segsum
// batch is sorted: accumulate runs in a register, atomic only on transitions.
__global__ __launch_bounds__(256) void k_segsum(
    const float* __restrict__ out, const int* __restrict__ batch,
    float* __restrict__ vn_temp, int n) {
  int c = threadIdx.x & 63;
  int grp = threadIdx.x >> 6;                // 4 row-groups of 64 rows
  int r0 = blockIdx.x * 256 + grp * 64;
  if (r0 >= n) return;
  int r1 = min(r0 + 64, n);
  float acc = 0.0f;
  int cur = batch[r0];
  for (int r = r0; r < r1; ++r) {
    int b = batch[r];
    if (b != cur) {
      atomicAdd(&vn_temp[cur * D_ + c], acc);
      acc = 0.0f;
      cur = b;
    }
    acc += out[(size_t)r * D_ + c];
  }
  atomicAdd(&vn_temp[cur * D_ + c], acc);
}

// -------------------------------------------------------------- K3: edges
// 16 threads per edge, float4 per thread (4 cols), 4 scalar f32 atomics.
__global__ __launch_bounds__(256) void k_edges(
    const float* __restrict__ out, const int* __restrict__ ei,
    const int* __restrict__ ea, const float* __restrict__ bond,
    float* __restrict__ aggr, int E_) {
  int idx = blockIdx.x * 256 + threadIdx.x;
  int e = idx >> 4;
  if (e >= E_) return;
  int cq = (idx & 15) * 4;
  int src = ei[e];
  int dst = ei[E_ + e];
  int a0 = ea[e * 3 + 0], a1 = ea[e * 3 + 1], a2 = ea[e * 3 + 2];
  const float4 v  = *(const float4*)(out + (size_t)src * D_ + cq);
  const float4 e0 = *(const float4*)(bond + (size_t)(0 * 5 + a0) * D_ + cq);
  const float4 e1 = *(const float4*)(bond + (size_t)(1 * 5 + a1) * D_ + cq);
  const float4 e2 = *(const float4*)(bond + (size_t)(2 * 5 + a2) * D_ + cq);
  float m0 = fmaxf(v.x + e0.x + e1.x + e2.x, 0.0f);
  float m1 = fmaxf(v.y + e0.y + e1.y + e2.y, 0.0f);
  float m2 = fmaxf(v.z + e0.z + e1.z + e2.z, 0.0f);
  float m3 = fmaxf(v.w + e0.w + e1.w + e2.w, 0.0f);
  float* dp = aggr + (size_t)dst * D_ + cq;
  atomicAdd(dp + 0, m0);
  atomicAdd(dp + 1, m1);
  atomicAdd(dp + 2, m2);
  atomicAdd(dp + 3, m3);
}

// ------------------------------------------------- K4/K5: fused 64->128->64 MLP
// One wave per 16-row tile. GEMM1 accumulates via v_wmma_f32_16x16x4_f32,
// epilogue (bias+BN+relu) lands in a padded LDS tile, GEMM2 reads transposed
// A-fragments back from LDS, second epilogue stores to global.
__global__ __launch_bounds__(128) void k_mlp(
    const float* __restrict__ A,     // [rows,64] primary input
    const float* __restrict__ Badd,  // [rows,64] addend (aggr) or nullptr
    const float* __restrict__ epsPtr,// GIN eps or nullptr
    const float* __restrict__ W1, const float* __restrict__ b1,
    const float* __restrict__ g1, const float* __restrict__ be1,
    const float* __restrict__ m1, const float* __restrict__ v1,
    const float* __restrict__ W2, const float* __restrict__ b2,
    const float* __restrict__ g2, const float* __restrict__ be2,
    const float* __restrict__ m2, const float* __restrict__ v2p,
    float* __restrict__ Out, int rows) {
  __shared__ float lds[4][16 * 132];           // 33 KB, per-wave 16x128 (+4 pad)
  const int wave = threadIdx.x >> 5;
  const int lane = threadIdx.x & 31;
  const int half = lane >> 4;                  // 0: K lo pair, 1: K hi pair
  const int l16 = lane & 15;
  const int tile = blockIdx.x * 4 + wave;
  const int row0 = tile * 16;
  if (row0 >= rows) return;                    // wave-uniform; EXEC stays ~0

  const float epsv = epsPtr ? (1.0f + epsPtr[0]) : 1.0f;

  // A-fragments per ISA 16x4 f32 layout: lane<16 holds K=k..k+1, lane>=16 K=k+2..k+3
  v2f afrag[16];
  {
    const int r = row0 + l16;
    const int kb = half * 2;
    const float* ap = A + (size_t)r * D_;
    const float* bp = Badd ? (Badd + (size_t)r * D_) : nullptr;
#pragma unroll
    for (int k = 0; k < 16; ++k) {
      int col = k * 4 + kb;
      v2f av = *(const v2f*)(ap + col);
      if (bp) {
        v2f bv = *(const v2f*)(bp + col);
        av = av * epsv + bv;                   // bb = (1+eps)*out + aggr
      }
      afrag[k] = av;
    }
  }

  float* l = &lds[wave][0];

  // GEMM1: [16x64] @ W1[64x128]
  for (int nt = 0; nt < 8; ++nt) {
    v8f acc = {};
    const int n = nt * 16 + l16;
#pragma unroll
    for (int k = 0; k < 16; ++k) {
      const int br = k * 4 + half * 2;
      v2f bf;
      bf.x = W1[br * 128 + n];
      bf.y = W1[(br + 1) * 128 + n];
      acc = wmma_f32_k4(afrag[k], bf, acc);
    }
    const float scale = g1[n] * rsqrtf(v1[n] + BN_EPS_);
    const float shift = (b1[n] - m1[n]) * scale + be1[n];
#pragma unroll
    for (int g = 0; g < 8; ++g) {
      const int m = g + half * 8;              // C layout: lanes 16-31 -> M+8
      float y = fmaxf(acc[g] * scale + shift, 0.0f);
      l[m * 132 + n] = y;
    }
  }

  asm volatile("s_wait_dscnt 0" ::: "memory"); // LDS store -> LDS load hazard

  // GEMM2: the 16x128 LDS tile @ W2[128x64]
  for (int nt = 0; nt < 4; ++nt) {
    v8f acc = {};
    const int n = nt * 16 + l16;
#pragma unroll
    for (int k = 0; k < 32; ++k) {
      const int kc = k * 4 + half * 2;
      v2f af = *(const v2f*)(l + l16 * 132 + kc);
      v2f bf;
      bf.x = W2[kc * 64 + n];
      bf.y = W2[(kc + 1) * 64 + n];
      acc = wmma_f32_k4(af, bf, acc);
    }
    const float scale = g2[n] * rsqrtf(v2p[n] + BN_EPS_);
    const float shift = (b2[n] - m2[n]) * scale + be2[n];
#pragma unroll
    for (int g = 0; g < 8; ++g) {
      const int m = g + half * 8;
      float y = fmaxf(acc[g] * scale + shift, 0.0f);
      Out[(size_t)(row0 + m) * D_ + n] = y;
    }
  }
}

// ------------------------------------------------------------------ launch
extern "C" void kernel_launch(void* const* d_in, const int* in_sizes, int n_in,
                              void* d_out, int out_size, void* d_ws,
                              size_t ws_size, hipStream_t stream) {
  (void)n_in; (void)out_size; (void)ws_size;
  const float* h0   = (const float*)d_in[0];
  const float* h1   = (const float*)d_in[1];
  const float* vn0  = (const float*)d_in[2];
  const float* vn1  = (const float*)d_in[3];
  const int*   batch = (const int*)d_in[4];
  const int*   edge_index = (const int*)d_in[5];
  const int*   edge_attr  = (const int*)d_in[6];
  const float* mw   = (const float*)d_in[7];
  const float* mwv  = (const float*)d_in[8];
  const float* bond = (const float*)d_in[9];
  const float* eps  = (const float*)d_in[10];
  const float* W1   = (const float*)d_in[11];
  const float* b1   = (const float*)d_in[12];
  const float* bn1_g = (const float*)d_in[13];
  const float* bn1_b = (const float*)d_in[14];
  const float* bn1_m = (const float*)d_in[15];
  const float* bn1_v = (const float*)d_in[16];
  const float* W2   = (const float*)d_in[17];
  const float* b2   = (const float*)d_in[18];
  const float* bno_g = (const float*)d_in[19];
  const float* bno_b = (const float*)d_in[20];
  const float* bno_m = (const float*)d_in[21];
  const float* bno_v = (const float*)d_in[22];
  const float* vW1  = (const float*)d_in[23];
  const float* vb1  = (const float*)d_in[24];
  const float* vbn1_g = (const float*)d_in[25];
  const float* vbn1_b = (const float*)d_in[26];
  const float* vbn1_m = (const float*)d_in[27];
  const float* vbn1_v = (const float*)d_in[28];
  const float* vW2  = (const float*)d_in[29];
  const float* vb2  = (const float*)d_in[30];
  const float* vbn2_g = (const float*)d_in[31];
  const float* vbn2_b = (const float*)d_in[32];
  const float* vbn2_m = (const float*)d_in[33];
  const float* vbn2_v = (const float*)d_in[34];

  const int N = in_sizes[4];          // batch: [N]
  const int E = in_sizes[5] / 2;      // edge_index: [2,E]
  const int G = in_sizes[2] / D_;     // vn0: [G,D]

  float* out     = (float*)d_ws;                 // [N,64]
  float* aggr    = out + (size_t)N * D_;         // [N,64]
  float* vn_temp = aggr + (size_t)N * D_;        // [G,64]

  float* h_out  = (float*)d_out;                 // [N,64]
  float* vn_out = h_out + (size_t)N * D_;        // [G,64]

  // K1: merge + vn broadcast, zero aggr, init vn_temp = vn_aggr
  {
    int total = N * D_;
    k_merge<<<(total + 255) / 256, 256, 0, stream>>>(
        h0, h1, vn0, vn1, batch, mw, mwv, out, aggr, vn_temp, N, G);
  }
  // K2: segment sum into vn_temp
  k_segsum<<<(N + 255) / 256, 256, 0, stream>>>(out, batch, vn_temp, N);
  // K3: edge message + scatter-add into aggr
  {
    long t = (long)E * 16;
    k_edges<<<(int)((t + 255) / 256), 256, 0, stream>>>(out, edge_index,
                                                        edge_attr, bond, aggr, E);
  }
  // K4: virtual-node MLP (depends on K2)
  {
    int tiles = (G + 15) / 16;
    k_mlp<<<(tiles + 3) / 4, 128, 0, stream>>>(
        vn_temp, nullptr, nullptr,
        vW1, vb1, vbn1_g, vbn1_b, vbn1_m, vbn1_v,
        vW2, vb2, vbn2_g, vbn2_b, vbn2_m, vbn2_v,
        vn_out, G);
  }
  // K5: node MLP on bb = (1+eps)*out + aggr (depends on K3)
  {
    int tiles = (N + 15) / 16;
    k_mlp<<<(tiles + 3) / 4, 128, 0, stream>>>(
        out, aggr, eps,
        W1, b1, bn1_g, bn1_b, bn1_m, bn1_v,
        W2, b2, bno_g, bno_b, bno_m, bno_v,
        h_out, N);
  }
}